// Alignment_42649025249392
// MI455X (gfx1250) — compile-verified
//
#include <hip/hip_runtime.h>
#include <hip/hip_bf16.h>
#include <stdint.h>

// ---------------------------------------------------------------------------
// Bahdanau attention for MI455X (gfx1250, wave32).
//   B=8, TQ=TY=400, Q_DIM=Y_DIM=256, ATT_DIM=128
// GEMM stages use V_WMMA_F32_16X16X4_F32 (fp32 end-to-end, matches reference).
// Score stage (tanh inside contraction) runs on VALU with v_exp/v_rcp, with
// tiles staged via GLOBAL_LOAD_ASYNC_TO_LDS_B128 (ASYNCcnt path) and read
// back as ds_load_b128.
// ---------------------------------------------------------------------------

typedef float v2f __attribute__((ext_vector_type(2)));
typedef float v4f __attribute__((ext_vector_type(4)));
typedef float v8f __attribute__((ext_vector_type(8)));

#define BB   8
#define TQ   400
#define TY   400
#define DDIM 256
#define ADIM 128
#define LPAD 132   // 128 + 4: 16B-aligned rows, conflict-free b128 LDS reads

// ---------------------------------------------------------------------------
// Kernel 1: projection  out[R,128] = X[R,256] * W[128,256]^T + bias
// One wave = one 16x16 C tile; 8 waves/block cover all 128 output columns.
// WMMA f32 16x16x4 fragment layout (ISA 7.12.2):
//   A (16x4): lane l -> row M = l%16, K pair = (l/16)*2 + {vgpr0,vgpr1}
//   B (4x16): lane l -> col N = l%16, K pair = (l/16)*2 + {vgpr0,vgpr1}
//   C (16x16): vgpr r -> row M = r + 8*(l/16), col N = l%16
// ---------------------------------------------------------------------------
__global__ void proj_kernel(const float* __restrict__ X,
                            const float* __restrict__ W,
                            const float* __restrict__ bias,
                            float* __restrict__ out)
{
    const int tile = blockIdx.x;            // row tile (16 rows)
    const int wave = threadIdx.x >> 5;      // 0..7 -> column tile
    const int lane = threadIdx.x & 31;
    const int lh   = lane >> 4;             // lane half: K-pair select
    const int lm   = lane & 15;             // M (for A) / N (for B,C)

    const int row0 = tile * 16;
    const int n0   = wave * 16;

    const float* __restrict__ Abase = X + (size_t)(row0 + lm) * DDIM + lh * 2;
    const float* __restrict__ Bbase = W + (size_t)(n0 + lm) * DDIM + lh * 2;

    v8f c = {};
#pragma unroll 4
    for (int k = 0; k < DDIM; k += 4) {
        v2f a = *(const v2f*)(Abase + k);
        v2f b = *(const v2f*)(Bbase + k);
        c = __builtin_amdgcn_wmma_f32_16x16x4_f32(
                false, a, false, b, (short)0, c, false, false);
    }

    const float bc = bias[n0 + lm];
#pragma unroll
    for (int r = 0; r < 8; ++r) {
        const int rowl = r + lh * 8;
        out[(size_t)(row0 + rowl) * ADIM + n0 + lm] = c[r] + bc;
    }
}

// ---------------------------------------------------------------------------
// Kernel 2: scores  att[b,q,t] = sum_a v[a]*tanh(qp[b,q,a]+yp[b,t,a]) + vb
// with mask (t >= n_wins[b] -> -inf).  16x16 (q,t) tile per 256-thread block.
// Tiles staged with async global->LDS B128 (no VGPR round trip), read back
// as 128-bit LDS loads. tanh(x) = 1 - 2*rcp(exp2(2*log2e*x)+1).
// ---------------------------------------------------------------------------
__global__ void score_kernel(const float* __restrict__ qp,
                             const float* __restrict__ yp,
                             const float* __restrict__ vw,
                             const float* __restrict__ vb,
                             const int*   __restrict__ nwins,
                             float* __restrict__ att)
{
    __shared__ __align__(16) float qs[16 * LPAD];
    __shared__ __align__(16) float ys[16 * LPAD];
    __shared__ __align__(16) float vs[ADIM];

    const int b   = blockIdx.y;
    const int qt  = blockIdx.x / (TY / 16);
    const int tt  = blockIdx.x % (TY / 16);
    const int tid = threadIdx.x;

    // Async-stage both 16x128 tiles: 512 16-byte chunks each, 2 per thread.
    // LDS aperture: low 32 bits of the flat shared address == LDS byte offset.
    const float* __restrict__ qsrc = qp + ((size_t)b * TQ + qt * 16) * ADIM;
    const float* __restrict__ ysrc = yp + ((size_t)b * TY + tt * 16) * ADIM;
#pragma unroll
    for (int cchunk = 0; cchunk < 2; ++cchunk) {
        const int ci = tid + cchunk * 256;       // 0..511
        const int r  = ci >> 5;                  // row 0..15
        const int g  = (ci & 31) << 2;           // col (floats), multiple of 4
        const uint32_t qdst = (uint32_t)(uintptr_t)&qs[r * LPAD + g];
        const uint32_t ydst = (uint32_t)(uintptr_t)&ys[r * LPAD + g];
        const float* qg = qsrc + r * ADIM + g;
        const float* yg = ysrc + r * ADIM + g;
        asm volatile("global_load_async_to_lds_b128 %0, %1, off"
                     :: "v"(qdst), "v"(qg) : "memory");
        asm volatile("global_load_async_to_lds_b128 %0, %1, off"
                     :: "v"(ydst), "v"(yg) : "memory");
    }
    if (tid < ADIM) vs[tid] = vw[tid];
    asm volatile("s_wait_asynccnt 0x0" ::: "memory");
    __syncthreads();

    const int qi = tid >> 4;
    const int ti = tid & 15;
    const v4f* __restrict__ qrow = (const v4f*)&qs[qi * LPAD];
    const v4f* __restrict__ yrow = (const v4f*)&ys[ti * LPAD];
    const v4f* __restrict__ vv4  = (const v4f*)vs;

    float s = vb[0];
#pragma unroll 4
    for (int a4 = 0; a4 < ADIM / 4; ++a4) {
        const v4f x = qrow[a4] + yrow[a4];       // ds_load_b128 x2
        const v4f w = vv4[a4];
#pragma unroll
        for (int j = 0; j < 4; ++j) {
            const float e  = __builtin_amdgcn_exp2f(x[j] * 2.885390081777927f);
            const float rr = __builtin_amdgcn_rcpf(e + 1.0f);
            s = fmaf(w[j], fmaf(-2.0f, rr, 1.0f), s);
        }
    }

    const int tg = tt * 16 + ti;
    const int qg2 = qt * 16 + qi;
    const float res = (tg < nwins[b]) ? s : -__builtin_inff();
    att[((size_t)b * TQ + qg2) * TY + tg] = res;
}

// ---------------------------------------------------------------------------
// Kernel 3: row softmax over Ty (400). One (b,q) row per 128-thread block.
// Masked entries are -inf -> exp() = 0; at least Ty/2 entries are valid.
// ---------------------------------------------------------------------------
__global__ void softmax_kernel(float* __restrict__ att)
{
    __shared__ float red[128];
    const int tid = threadIdx.x;
    float* __restrict__ p = att + (size_t)blockIdx.x * TY;

    float m = -__builtin_inff();
    for (int t = tid; t < TY; t += 128) m = fmaxf(m, p[t]);
    red[tid] = m;
    __syncthreads();
#pragma unroll
    for (int s = 64; s > 0; s >>= 1) {
        if (tid < s) red[tid] = fmaxf(red[tid], red[tid + s]);
        __syncthreads();
    }
    m = red[0];
    __syncthreads();

    float sum = 0.0f;
    for (int t = tid; t < TY; t += 128) {
        const float e = __builtin_amdgcn_exp2f((p[t] - m) * 1.4426950408889634f);
        p[t] = e;
        sum += e;
    }
    red[tid] = sum;
    __syncthreads();
#pragma unroll
    for (int s = 64; s > 0; s >>= 1) {
        if (tid < s) red[tid] += red[tid + s];
        __syncthreads();
    }
    const float inv = 1.0f / red[0];
    for (int t = tid; t < TY; t += 128) p[t] *= inv;
}

// ---------------------------------------------------------------------------
// Kernel 4: out[b,q,d] = sum_t P[b,q,t] * y[b,t,d]   (WMMA f32 16x16x4)
// gridDim.x = B * TQ/16, gridDim.y = 2; wave w handles d-tile (y*8 + w).
// ---------------------------------------------------------------------------
__global__ void apply_kernel(const float* __restrict__ att,
                             const float* __restrict__ y,
                             float* __restrict__ out)
{
    const int idx  = blockIdx.x;
    const int b    = idx / (TQ / 16);
    const int qt   = idx % (TQ / 16);
    const int wave = threadIdx.x >> 5;
    const int lane = threadIdx.x & 31;
    const int lh   = lane >> 4;
    const int lm   = lane & 15;

    const int d0 = (blockIdx.y * 8 + wave) * 16;

    const float* __restrict__ Abase =
        att + ((size_t)b * TQ + qt * 16 + lm) * TY + lh * 2;
    const float* __restrict__ Ybase =
        y + (size_t)b * TY * DDIM + d0 + lm;

    v8f c = {};
#pragma unroll 2
    for (int t = 0; t < TY; t += 4) {
        v2f a = *(const v2f*)(Abase + t);
        v2f bb;
        const int tr = t + lh * 2;
        bb.x = Ybase[(size_t)tr * DDIM];
        bb.y = Ybase[(size_t)(tr + 1) * DDIM];
        c = __builtin_amdgcn_wmma_f32_16x16x4_f32(
                false, a, false, bb, (short)0, c, false, false);
    }

#pragma unroll
    for (int r = 0; r < 8; ++r) {
        const int rowl = r + lh * 8;
        out[((size_t)b * TQ + qt * 16 + rowl) * DDIM + d0 + lm] = c[r];
    }
}

// ---------------------------------------------------------------------------
// Launcher
// ---------------------------------------------------------------------------
extern "C" void kernel_launch(void* const* d_in, const int* in_sizes, int n_in,
                              void* d_out, int out_size, void* d_ws, size_t ws_size,
                              hipStream_t stream)
{
    (void)in_sizes; (void)n_in; (void)out_size; (void)ws_size;

    const float* query = (const float*)d_in[0];   // [B,TQ,256]
    const float* yin   = (const float*)d_in[1];   // [B,TY,256]
    const float* Wq_w  = (const float*)d_in[2];   // [128,256]
    const float* Wq_b  = (const float*)d_in[3];   // [128]
    const float* Wy_w  = (const float*)d_in[4];   // [128,256]
    const float* Wy_b  = (const float*)d_in[5];   // [128]
    const float* v_w   = (const float*)d_in[6];   // [1,128]
    const float* v_b   = (const float*)d_in[7];   // [1]
    const int*   nwins = (const int*)d_in[8];     // [B]
    float*       out   = (float*)d_out;           // [B,TQ,256]

    float* qp  = (float*)d_ws;                       // [B*TQ,128]
    float* yp  = qp + (size_t)BB * TQ * ADIM;        // [B*TY,128]
    float* att = yp + (size_t)BB * TY * ADIM;        // [B*TQ,TY]

    // Projections (WMMA f32): 3200 rows -> 200 row tiles, 8 waves = 128 cols
    proj_kernel<<<(BB * TQ) / 16, 256, 0, stream>>>(query, Wq_w, Wq_b, qp);
    proj_kernel<<<(BB * TY) / 16, 256, 0, stream>>>(yin,   Wy_w, Wy_b, yp);

    // Scores + mask: 25x25 tiles per batch, async-LDS staged
    score_kernel<<<dim3((TQ / 16) * (TY / 16), BB), 256, 0, stream>>>(
        qp, yp, v_w, v_b, nwins, att);

    // Softmax over Ty
    softmax_kernel<<<BB * TQ, 128, 0, stream>>>(att);

    // Attention apply (WMMA f32)
    apply_kernel<<<dim3((BB * TQ) / 16, 2), 256, 0, stream>>>(att, yin, out);
}